// WindowedAttentionBlock_2705829396775
// MI455X (gfx1250) — compile-verified
//
#include <hip/hip_runtime.h>
#include <hip/hip_bf16.h>

// ---------------------------------------------------------------------------
// WindowedAttentionBlock for MI455X (gfx1250), fp32 end-to-end.
// C=512, HEADS=16, hd=32, WIN=(2,2,2,2) -> Nwin=16, PATCHES=(2,2,64,128),
// Nb=2 -> 65536 tokens, 4096 windows (nt=nl=1, nh=32, nw=64).
// All GEMMs: V_WMMA_F32_16X16X4_F32 (native fp32 matrix op, wave32).
// Each block processes 32 tokens (2 windows / 2 M-tiles) so every weight
// B-fragment feeds two WMMAs -> half the L2 weight traffic of an M=16 design.
// ---------------------------------------------------------------------------

typedef __attribute__((ext_vector_type(2))) float v2f;
typedef __attribute__((ext_vector_type(8))) float v8f;

#define XS 516   // 512 + pad (stride mod 64 == 4 -> conflict-free A-frag reads)
#define QS 1540  // 1536 + pad

__device__ __forceinline__ v8f wmma4(v2f a, v2f b, v8f c) {
  // D = A(16x4 f32) * B(4x16 f32) + C(16x16 f32)
  return __builtin_amdgcn_wmma_f32_16x16x4_f32(
      /*neg_a=*/false, a, /*neg_b=*/false, b,
      /*c_mod=*/(short)0, c, /*reuse_a=*/false, /*reuse_b=*/false);
}

#define ZERO8 v8f{0.f,0.f,0.f,0.f,0.f,0.f,0.f,0.f}

// ===========================================================================
// Kernel A: window gather + QKV GEMM + biased attention + proj + LN1 + resid
// grid = 2048 (b:2, lon-window-pair:32, lat-window:32), block = 512 (16 waves)
// Two lon-adjacent windows per block -> 32 token rows (2 WMMA M-tiles).
// ===========================================================================
__global__ __launch_bounds__(512) void winattn_kernel(
    const float* __restrict__ x,          // (2,512,2,2,64,128)
    const float* __restrict__ qkv_w,      // (512,1536)
    const float* __restrict__ proj_w,     // (512,512)
    const float* __restrict__ proj_b,     // (512)
    const float* __restrict__ bias_table, // (192,32,16)
    const float* __restrict__ n1g, const float* __restrict__ n1b,
    float* __restrict__ ws)               // xt, token-major (65536,512)
{
  extern __shared__ float smem[];
  float* lx = smem;            // 32 x XS : window tokens (also residual x0)
  float* lq = smem + 32 * XS;  // 32 x QS : QKV -> (O | Y) scratch
  float* mu = lq + 32 * QS;    // 32
  float* rs = mu + 32;         // 32

  const int tid  = threadIdx.x;
  const int wave = tid >> 5;   // 0..15
  const int lane = tid & 31;
  const int half = lane >> 4;
  const int lr   = lane & 15;

  const int bb  = blockIdx.x >> 10;        // batch
  const int r   = blockIdx.x & 1023;
  const int iwp = r >> 5;                  // longitude window PAIR (0..31)
  const int ihw = r & 31;                  // latitude window = bias "type"

  // ---- gather 2 windows (32 tokens x 512 ch) into LDS ----------------------
  for (int f = tid; f < 32 * 512; f += 512) {
    const int nn = f >> 9, c = f & 511;
    const int win = nn >> 4, n = nn & 15;
    const int tt = n >> 3, tl = (n >> 2) & 1, th = (n >> 1) & 1, tw = n & 1;
    const int sp = tt * 16384 + tl * 8192 + (ihw * 2 + th) * 128
                 + (iwp * 4 + win * 2 + tw);
    lx[nn * XS + c] = x[((size_t)bb * 512 + c) * 32768 + sp];
  }
  __syncthreads();

  // ---- QKV: (32x512)@(512x1536); wave owns 6 N-tiles x 2 M-tiles -----------
  {
    v8f acc0[6], acc1[6];
    for (int j = 0; j < 6; ++j) { acc0[j] = ZERO8; acc1[j] = ZERO8; }
    for (int kk = 0; kk < 512; kk += 4) {
      const int ka = kk + 2 * half;
      v2f a0; a0.x = lx[lr * XS + ka];        a0.y = lx[lr * XS + ka + 1];
      v2f a1; a1.x = lx[(16 + lr) * XS + ka]; a1.y = lx[(16 + lr) * XS + ka + 1];
      __builtin_prefetch(&qkv_w[(size_t)(ka + 8) * 1536 + wave * 96 + lr], 0, 0);
#pragma unroll
      for (int j = 0; j < 6; ++j) {
        const int col = (wave * 6 + j) * 16 + lr;
        v2f b; b.x = qkv_w[(size_t)ka * 1536 + col];
               b.y = qkv_w[(size_t)(ka + 1) * 1536 + col];
        acc0[j] = wmma4(a0, b, acc0[j]);   // B fragment reused for both M-tiles
        acc1[j] = wmma4(a1, b, acc1[j]);
      }
    }
    for (int j = 0; j < 6; ++j) {
      const int n0 = (wave * 6 + j) * 16;
#pragma unroll
      for (int e = 0; e < 8; ++e) {
        lq[(e + 8 * half) * QS + n0 + lr]        = acc0[j][e];
        lq[(16 + e + 8 * half) * QS + n0 + lr]   = acc1[j][e];
      }
    }
  }
  __syncthreads();

  // ---- attention: one thread per (window, head, query-row) -----------------
  {
    const int win = tid >> 8;          // 0,1
    const int hh  = (tid >> 4) & 15;   // head
    const int n   = tid & 15;          // query row
    const int nr  = win * 16 + n;      // token row in LDS
    const float scale = 0.17677669529663688f; // 32^-0.5
    float qv[32];
#pragma unroll
    for (int d = 0; d < 32; ++d) qv[d] = lq[nr * QS + hh * 32 + d] * scale;

    const int nt_ = n >> 3, nl_ = (n >> 2) & 1, nh_ = (n >> 1) & 1, nw_ = n & 1;
    float sc[16];
    float mx = -1e30f;
    for (int m = 0; m < 16; ++m) {
      const int mr = win * 16 + m;
      float s = 0.f;
#pragma unroll
      for (int d = 0; d < 32; ++d) s += qv[d] * lq[mr * QS + 512 + hh * 32 + d];
      const int mt = m >> 3, ml = (m >> 2) & 1, mh = (m >> 1) & 1, mw = m & 1;
      const int pos = 48 * (nt_ + 2 * mt) + 12 * (nl_ + 2 * ml)
                    + 3 * (nh_ + 2 * mh) + (nw_ - mw + 1);
      s += bias_table[(pos * 32 + ihw) * 16 + hh];
      sc[m] = s; mx = fmaxf(mx, s);
    }
    float sum = 0.f;
    for (int m = 0; m < 16; ++m) { sc[m] = __expf(sc[m] - mx); sum += sc[m]; }
    const float inv = 1.f / sum;
    for (int d = 0; d < 32; ++d) {
      float o = 0.f;
#pragma unroll
      for (int m = 0; m < 16; ++m) o += sc[m] * lq[(win * 16 + m) * QS + 1024 + hh * 32 + d];
      lq[nr * QS + hh * 32 + d] = o * inv;  // O overwrites this thread's Q slice
    }
  }
  __syncthreads();

  // ---- proj: (32x512)@(512x512); wave owns 2 N-tiles x 2 M-tiles -----------
  {
    v8f acc0[2], acc1[2];
    for (int j = 0; j < 2; ++j) { acc0[j] = ZERO8; acc1[j] = ZERO8; }
    for (int kk = 0; kk < 512; kk += 4) {
      const int ka = kk + 2 * half;
      v2f a0; a0.x = lq[lr * QS + ka];        a0.y = lq[lr * QS + ka + 1];
      v2f a1; a1.x = lq[(16 + lr) * QS + ka]; a1.y = lq[(16 + lr) * QS + ka + 1];
#pragma unroll
      for (int j = 0; j < 2; ++j) {
        const int col = (wave * 2 + j) * 16 + lr;
        v2f b; b.x = proj_w[(size_t)ka * 512 + col];
               b.y = proj_w[(size_t)(ka + 1) * 512 + col];
        acc0[j] = wmma4(a0, b, acc0[j]);
        acc1[j] = wmma4(a1, b, acc1[j]);
      }
    }
    for (int j = 0; j < 2; ++j) {
      const int n0 = (wave * 2 + j) * 16;
      const float pb = proj_b[n0 + lr];
#pragma unroll
      for (int e = 0; e < 8; ++e) {
        lq[(e + 8 * half) * QS + 512 + n0 + lr]      = acc0[j][e] + pb; // Y in K slot
        lq[(16 + e + 8 * half) * QS + 512 + n0 + lr] = acc1[j][e] + pb;
      }
    }
  }
  __syncthreads();

  // ---- LN1 stats: wave w -> tokens 2w, 2w+1 (32 tokens, 16 waves) ----------
  for (int t2 = 0; t2 < 2; ++t2) {
    const int tok = wave * 2 + t2;
    float s = 0.f, s2 = 0.f;
    for (int c = lane; c < 512; c += 32) {
      const float v = lq[tok * QS + 512 + c];
      s += v; s2 += v * v;
    }
    for (int off = 16; off >= 1; off >>= 1) {
      s  += __shfl_xor(s,  off, 32);
      s2 += __shfl_xor(s2, off, 32);
    }
    if (lane == 0) {
      const float m = s * (1.f / 512.f);
      mu[tok] = m;
      rs[tok] = rsqrtf(s2 * (1.f / 512.f) - m * m + 1e-6f);
    }
  }
  __syncthreads();

  // ---- xt = x0 + LN1(y); token-major store to workspace --------------------
  for (int f = tid; f < 32 * 512; f += 512) {
    const int nn = f >> 9, c = f & 511;
    const int win = nn >> 4, n = nn & 15;
    const int tt = n >> 3, tl = (n >> 2) & 1, th = (n >> 1) & 1, tw = n & 1;
    const int sp = tt * 16384 + tl * 8192 + (ihw * 2 + th) * 128
                 + (iwp * 4 + win * 2 + tw);
    const float v = lx[nn * XS + c]
                  + (lq[nn * QS + 512 + c] - mu[nn]) * rs[nn] * n1g[c] + n1b[c];
    ws[((size_t)bb * 32768 + sp) * 512 + c] = v;
  }
}

// ===========================================================================
// Kernel B: MLP (hidden tiled 4x512, fc2 partials in registers) + LN2 + resid
// grid = 2048 (32 consecutive tokens each), block = 512 (16 waves)
// ===========================================================================
__global__ __launch_bounds__(512) void mlp_kernel(
    const float* __restrict__ xt,         // (65536,512) token-major
    const float* __restrict__ fc1_w,      // (512,2048)
    const float* __restrict__ fc1_b,      // (2048)
    const float* __restrict__ fc2_w,      // (2048,512)
    const float* __restrict__ fc2_b,      // (512)
    const float* __restrict__ n2g, const float* __restrict__ n2b,
    float* __restrict__ out)              // (2,512,32768)
{
  extern __shared__ float smem[];
  float* lx = smem;            // 32 x XS : xt tile (residual)
  float* lh = smem + 32 * XS;  // 32 x XS : hidden chunk, then y
  float* mu = lh + 32 * XS;
  float* rs = mu + 32;

  const int tid  = threadIdx.x;
  const int wave = tid >> 5;   // 0..15
  const int lane = tid & 31;
  const int half = lane >> 4;
  const int lr   = lane & 15;
  const size_t g0 = (size_t)blockIdx.x * 32;

  for (int f = tid; f < 32 * 512; f += 512) {
    const int n = f >> 9, c = f & 511;
    lx[n * XS + c] = xt[(g0 + n) * 512 + c];
  }
  __syncthreads();

  v8f oacc0[2], oacc1[2];
  for (int j = 0; j < 2; ++j) { oacc0[j] = ZERO8; oacc1[j] = ZERO8; }

  for (int ch = 0; ch < 4; ++ch) {
    const int hc0 = ch * 512;
    // -- h_chunk = gelu(xt @ fc1[:, hc0:hc0+512] + b1) ------------------------
    v8f hacc0[2], hacc1[2];
    for (int j = 0; j < 2; ++j) { hacc0[j] = ZERO8; hacc1[j] = ZERO8; }
    for (int kk = 0; kk < 512; kk += 4) {
      const int ka = kk + 2 * half;
      v2f a0; a0.x = lx[lr * XS + ka];        a0.y = lx[lr * XS + ka + 1];
      v2f a1; a1.x = lx[(16 + lr) * XS + ka]; a1.y = lx[(16 + lr) * XS + ka + 1];
      __builtin_prefetch(&fc1_w[(size_t)(ka + 8) * 2048 + hc0 + wave * 32 + lr], 0, 0);
#pragma unroll
      for (int j = 0; j < 2; ++j) {
        const int col = hc0 + (wave * 2 + j) * 16 + lr;
        v2f b; b.x = fc1_w[(size_t)ka * 2048 + col];
               b.y = fc1_w[(size_t)(ka + 1) * 2048 + col];
        hacc0[j] = wmma4(a0, b, hacc0[j]);
        hacc1[j] = wmma4(a1, b, hacc1[j]);
      }
    }
    for (int j = 0; j < 2; ++j) {
      const int n0 = (wave * 2 + j) * 16;
      const float b1 = fc1_b[hc0 + n0 + lr];
#pragma unroll
      for (int e = 0; e < 8; ++e) {
        const float v0 = hacc0[j][e] + b1;
        const float v1 = hacc1[j][e] + b1;
        lh[(e + 8 * half) * XS + n0 + lr] =
            0.5f * v0 * (1.f + erff(v0 * 0.70710678118654752f)); // exact GELU
        lh[(16 + e + 8 * half) * XS + n0 + lr] =
            0.5f * v1 * (1.f + erff(v1 * 0.70710678118654752f));
      }
    }
    __syncthreads();
    // -- out += h_chunk @ fc2[hc0:hc0+512, :] --------------------------------
    for (int kk = 0; kk < 512; kk += 4) {
      const int ka = kk + 2 * half;
      v2f a0; a0.x = lh[lr * XS + ka];        a0.y = lh[lr * XS + ka + 1];
      v2f a1; a1.x = lh[(16 + lr) * XS + ka]; a1.y = lh[(16 + lr) * XS + ka + 1];
#pragma unroll
      for (int j = 0; j < 2; ++j) {
        const int col = (wave * 2 + j) * 16 + lr;
        v2f b; b.x = fc2_w[(size_t)(hc0 + ka) * 512 + col];
               b.y = fc2_w[(size_t)(hc0 + ka + 1) * 512 + col];
        oacc0[j] = wmma4(a0, b, oacc0[j]);
        oacc1[j] = wmma4(a1, b, oacc1[j]);
      }
    }
    __syncthreads();  // lh is reused next chunk
  }

  // ---- y = out + b2 into lh ------------------------------------------------
  for (int j = 0; j < 2; ++j) {
    const int n0 = (wave * 2 + j) * 16;
    const float b2 = fc2_b[n0 + lr];
#pragma unroll
    for (int e = 0; e < 8; ++e) {
      lh[(e + 8 * half) * XS + n0 + lr]      = oacc0[j][e] + b2;
      lh[(16 + e + 8 * half) * XS + n0 + lr] = oacc1[j][e] + b2;
    }
  }
  __syncthreads();

  // ---- LN2 stats -----------------------------------------------------------
  for (int t2 = 0; t2 < 2; ++t2) {
    const int tok = wave * 2 + t2;
    float s = 0.f, s2 = 0.f;
    for (int c = lane; c < 512; c += 32) {
      const float v = lh[tok * XS + c];
      s += v; s2 += v * v;
    }
    for (int off = 16; off >= 1; off >>= 1) {
      s  += __shfl_xor(s,  off, 32);
      s2 += __shfl_xor(s2, off, 32);
    }
    if (lane == 0) {
      const float m = s * (1.f / 512.f);
      mu[tok] = m;
      rs[tok] = rsqrtf(s2 * (1.f / 512.f) - m * m + 1e-6f);
    }
  }
  __syncthreads();

  // ---- out = xt + LN2(y), transposed store to (N,C,S) ----------------------
  const int b = (int)(g0 >> 15);
  const int sbase = (int)(g0 & 32767);
  for (int f = tid; f < 32 * 512; f += 512) {
    const int n = f & 31, c = f >> 5;  // 32 consecutive tokens per c -> coalesced
    const float v = lx[n * XS + c]
                  + (lh[n * XS + c] - mu[n]) * rs[n] * n2g[c] + n2b[c];
    out[((size_t)b * 512 + c) * 32768 + sbase + n] = v;
  }
}

// ===========================================================================
extern "C" void kernel_launch(void* const* d_in, const int* in_sizes, int n_in,
                              void* d_out, int out_size, void* d_ws, size_t ws_size,
                              hipStream_t stream) {
  (void)in_sizes; (void)n_in; (void)out_size; (void)ws_size;
  const float* x          = (const float*)d_in[0];
  const float* qkv_w      = (const float*)d_in[1];
  const float* proj_w     = (const float*)d_in[2];
  const float* proj_b     = (const float*)d_in[3];
  const float* bias_table = (const float*)d_in[4];
  const float* n1g        = (const float*)d_in[5];
  const float* n1b        = (const float*)d_in[6];
  const float* n2g        = (const float*)d_in[7];
  const float* n2b        = (const float*)d_in[8];
  const float* fc1_w      = (const float*)d_in[9];
  const float* fc1_b      = (const float*)d_in[10];
  const float* fc2_w      = (const float*)d_in[11];
  const float* fc2_b      = (const float*)d_in[12];
  float* out = (float*)d_out;
  float* ws  = (float*)d_ws;  // 65536*512 floats = 128 MiB xt scratch

  const size_t ldsA = (size_t)(32 * XS + 32 * QS + 64) * sizeof(float); // ~257 KB
  const size_t ldsB = (size_t)(32 * XS + 32 * XS + 64) * sizeof(float); // ~129 KB
  hipFuncSetAttribute(reinterpret_cast<const void*>(winattn_kernel),
                      hipFuncAttributeMaxDynamicSharedMemorySize, (int)ldsA);
  hipFuncSetAttribute(reinterpret_cast<const void*>(mlp_kernel),
                      hipFuncAttributeMaxDynamicSharedMemorySize, (int)ldsB);

  winattn_kernel<<<2048, 512, ldsA, stream>>>(x, qkv_w, proj_w, proj_b,
                                              bias_table, n1g, n1b, ws);
  mlp_kernel<<<2048, 512, ldsB, stream>>>(ws, fc1_w, fc1_b, fc2_w, fc2_b,
                                          n2g, n2b, out);
}